// GHMC_77043123355671
// MI455X (gfx1250) — compile-verified
//
#include <hip/hip_runtime.h>
#include <hip/hip_bf16.h>
#include <cstdint>
#include <math.h>

// ---------------------------------------------------------------------------
// GHM-C loss for MI455X (gfx1250), compile-only optimized:
//   pass1: single 256MB streaming pass (async global->LDS double buffer),
//          wave32 shuffle softmax reductions, LDS histogram
//   pass2: tiny kernels over the 2MB per-row summary
//   final: deterministic reduction via V_WMMA_F32_16X16X4_F32
// Roofline: 256MB @ 23.3TB/s ~= 11us floor; everything else is noise.
// ---------------------------------------------------------------------------

typedef __attribute__((ext_vector_type(2))) float v2f;
typedef __attribute__((ext_vector_type(8))) float v8f;

#define HIST_MAX   1024
#define WPB        8        // waves per block (256 threads, wave32)
#define P1_BLOCKS  4096
#define DOT_BLOCKS 512

// ---- CDNA5 async global->LDS path (guarded; falls back to plain loads) ----
#if __has_builtin(__builtin_amdgcn_global_load_async_to_lds_b128)
#define HAVE_ASYNC_LDS 1
#else
#define HAVE_ASYNC_LDS 0
#endif

#if __has_builtin(__builtin_amdgcn_s_wait_asynccnt)
#define WAIT_ASYNCCNT(n) __builtin_amdgcn_s_wait_asynccnt(n)
#else
#define WAIT_ASYNCCNT(n) asm volatile("s_wait_asynccnt %0" :: "n"(n) : "memory")
#endif

#if HAVE_ASYNC_LDS
// builtin signature (from clang diagnostic): b128 variant takes
//   (v4i addrspace(1)*, v4i addrspace(3)*, imm offset, imm cpol)
typedef int v4i __attribute__((vector_size(4 * sizeof(int))));
typedef __attribute__((address_space(1))) v4i* gv4p;
typedef __attribute__((address_space(3))) v4i* lv4p;

static __device__ __forceinline__ gv4p to_gbl(const void* p) {
    return (gv4p)(uintptr_t)p;
}
static __device__ __forceinline__ lv4p to_lds(void* p) {
    // generic LDS address: low 32 bits are the LDS byte offset (ISA 10.2)
    return (lv4p)(uintptr_t)(uint32_t)(uintptr_t)p;
}
#endif

// stage one row-slice (this lane's 2x16B) from global into LDS
static __device__ __forceinline__ void stage_row(const float* __restrict__ src,
                                                 float* dst) {
#if HAVE_ASYNC_LDS
    __builtin_amdgcn_global_load_async_to_lds_b128(to_gbl(src),       to_lds(dst),       0, 0);
    __builtin_amdgcn_global_load_async_to_lds_b128(to_gbl(src + 128), to_lds(dst + 128), 0, 0);
#else
    *(float4*)dst         = *(const float4*)src;
    *(float4*)(dst + 128) = *(const float4*)(src + 128);
#endif
}

// ---------------------------------------------------------------------------
// kernel 0: compute bins from epoch/batch device scalars; zero global hist
// ---------------------------------------------------------------------------
__global__ void ghmc_setup(const int* __restrict__ epoch, const int* __restrict__ batch,
                           int* __restrict__ meta, int* __restrict__ hist) {
    const int t = threadIdx.x;
    if (t == 0) {
        const double e = (double)epoch[0];
        const double b = (double)batch[0];
        const double x = ((e - 1.0) * 100.0 + b + 1.0) / 20000.0;
        int bins = (int)floor(pow(40.0, x));   // matches python int(40**x)
        if (bins < 1) bins = 1;
        if (bins > HIST_MAX) bins = HIST_MAX;
        meta[0] = bins;
    }
    for (int i = t; i < HIST_MAX; i += 256) hist[i] = 0;
}

// ---------------------------------------------------------------------------
// kernel 1: streaming pass. One wave per row (C == 256 fixed by reference).
// Produces lp_t[row], bin[row], and the bin histogram.
// ---------------------------------------------------------------------------
__global__ __launch_bounds__(256) void ghmc_pass1(
    const float* __restrict__ predo, const int* __restrict__ target,
    const int* __restrict__ meta, int* __restrict__ ghist,
    float* __restrict__ lp_out, int* __restrict__ bin_out, int N) {

    __shared__ __align__(16) float sbuf[WPB][2][256];  // 16 KB double buffer
    __shared__ int shist[HIST_MAX];                    // 4 KB

    const int tid = threadIdx.x;
    for (int i = tid; i < HIST_MAX; i += 256) shist[i] = 0;
    __syncthreads();

    const int   bins  = meta[0];
    const float binsF = (float)bins;
    const int   lane  = tid & 31;
    const int   w     = tid >> 5;
    const int   wid   = blockIdx.x * WPB + w;
    const int   nwv   = gridDim.x * WPB;

    int row = wid;
    int parity = 0;
    if (row < N)
        stage_row(predo + ((size_t)row << 8) + lane * 4, &sbuf[w][0][lane * 4]);

    for (; row < N; row += nwv) {
        const int nxt = row + nwv;
        if (nxt < N)
            stage_row(predo + ((size_t)nxt << 8) + lane * 4, &sbuf[w][parity ^ 1][lane * 4]);
#if HAVE_ASYNC_LDS
        if (nxt < N) { WAIT_ASYNCCNT(2); } else { WAIT_ASYNCCNT(0); }
        asm volatile("" ::: "memory");   // keep LDS reads below the wait
#endif
        // lane covers columns [lane*4, lane*4+4) and [128+lane*4, ...): all
        // 64 LDS banks hit exactly once per b128 phase (conflict-free).
        const float4 x0 = *(const float4*)&sbuf[w][parity][lane * 4];
        const float4 x1 = *(const float4*)&sbuf[w][parity][128 + lane * 4];

        float mx = fmaxf(fmaxf(fmaxf(x0.x, x0.y), fmaxf(x0.z, x0.w)),
                         fmaxf(fmaxf(x1.x, x1.y), fmaxf(x1.z, x1.w)));
        #pragma unroll
        for (int o = 16; o >= 1; o >>= 1) mx = fmaxf(mx, __shfl_xor(mx, o, 32));

        float se = __expf(x0.x - mx) + __expf(x0.y - mx) + __expf(x0.z - mx) + __expf(x0.w - mx)
                 + __expf(x1.x - mx) + __expf(x1.y - mx) + __expf(x1.z - mx) + __expf(x1.w - mx);
        #pragma unroll
        for (int o = 16; o >= 1; o >>= 1) se += __shfl_xor(se, o, 32);

        const int   tcol = target[row];                       // broadcast load
        const float xt   = predo[((size_t)row << 8) + tcol];  // L2-hot reload
        const float lpt  = (xt - mx) - __logf(se);            // log p(target)
        const float p    = __expf(lpt);
        float g = 1.0f - p;                                   // |p_t - 1|
        g = fminf(fmaxf(g, 0.0f), 1.0f);
        int bi = (int)(g * binsF);
        bi = (bi < bins) ? bi : (bins - 1);                   // clip floor bucket

        if (lane == 0) {
            lp_out[row]  = lpt;
            bin_out[row] = bi;
            atomicAdd(&shist[bi], 1);
        }
        parity ^= 1;
    }

    __syncthreads();
    for (int i = tid; i < bins; i += 256)
        if (shist[i] != 0) atomicAdd(&ghist[i], shist[i]);
}

// ---------------------------------------------------------------------------
// kernel 2: per-bin weights  w[b] = (count>0) ? 1/(count * n_nonempty) : 0
// ---------------------------------------------------------------------------
__global__ void ghmc_weights(const int* __restrict__ hist, const int* __restrict__ meta,
                             float* __restrict__ wbin) {
    __shared__ int nnz_s;
    const int bins = meta[0];
    const int t = threadIdx.x;
    if (t == 0) nnz_s = 0;
    __syncthreads();
    for (int i = t; i < bins; i += 256)
        if (hist[i] > 0) atomicAdd(&nnz_s, 1);
    __syncthreads();
    const float nnz = fmaxf((float)nnz_s, 1.0f);
    for (int i = t; i < bins; i += 256) {
        const int c = hist[i];
        wbin[i] = (c > 0) ? 1.0f / ((float)c * nnz) : 0.0f;
    }
}

// ---------------------------------------------------------------------------
// kernel 3: weighted CE partial sums (deterministic in-block tree reduce)
// ---------------------------------------------------------------------------
__global__ __launch_bounds__(256) void ghmc_dot(
    const float* __restrict__ lp, const int* __restrict__ bidx,
    const float* __restrict__ wbin, float* __restrict__ partials, int N) {
    __shared__ float red[256];
    float acc = 0.0f;
    for (int i = blockIdx.x * 256 + threadIdx.x; i < N; i += gridDim.x * 256)
        acc += (-lp[i]) * wbin[bidx[i]];
    red[threadIdx.x] = acc;
    __syncthreads();
    #pragma unroll
    for (int s = 128; s > 0; s >>= 1) {
        if (threadIdx.x < s) red[threadIdx.x] += red[threadIdx.x + s];
        __syncthreads();
    }
    if (threadIdx.x == 0) partials[blockIdx.x] = red[0];
}

// ---------------------------------------------------------------------------
// kernel 4: sum 512 partials with 8 chained V_WMMA_F32_16X16X4_F32 ops.
// B = ones => D[m,n] = rowsum(A[m,:]) accumulated into C. One wave, EXEC all-1.
// A layout (32-bit 16x4): lane l -> M=l&15; VGPR0/1 hold K=(l>>4)*2, +1.
// C/D layout: lane 0 holds M=0..7 of N=0; lane 16 holds M=8..15 of N=0.
// ---------------------------------------------------------------------------
__global__ void ghmc_final(const float* __restrict__ partials, float* __restrict__ out) {
    const int lane = threadIdx.x;              // launched with exactly 32 threads
    const int m    = lane & 15;
    const int koff = (lane >> 4) * 2;
#if __has_builtin(__builtin_amdgcn_wmma_f32_16x16x4_f32)
    v8f c = {0.f, 0.f, 0.f, 0.f, 0.f, 0.f, 0.f, 0.f};
    v2f b; b.x = 1.0f; b.y = 1.0f;
    #pragma unroll
    for (int t = 0; t < DOT_BLOCKS / 64; ++t) {
        const int idx = t * 64 + m * 4 + koff;  // x[idx] = A[m][k], k = idx&3
        v2f a; a.x = partials[idx]; a.y = partials[idx + 1];
        c = __builtin_amdgcn_wmma_f32_16x16x4_f32(
                /*neg_a=*/false, a, /*neg_b=*/false, b,
                /*c_mod=*/(short)0, c, /*reuse_a=*/false, /*reuse_b=*/false);
    }
    float s = c[0] + c[1] + c[2] + c[3] + c[4] + c[5] + c[6] + c[7]; // M=0..7 / 8..15 of col N
    s += __shfl_xor(s, 16, 32);                 // combine lane0 + lane16 halves
#else
    float s = 0.0f;
    for (int t = 0; t < DOT_BLOCKS / 64; ++t) {
        const int idx = t * 64 + m * 4 + koff;
        s += partials[idx] + partials[idx + 1];
    }
    #pragma unroll
    for (int o = 16; o >= 1; o >>= 1) s += __shfl_xor(s, o, 32);
#endif
    if (lane == 0) out[0] = s;                  // * LOSS_WEIGHT (1.0)
}

// ---------------------------------------------------------------------------
extern "C" void kernel_launch(void* const* d_in, const int* in_sizes, int n_in,
                              void* d_out, int out_size, void* d_ws, size_t ws_size,
                              hipStream_t stream) {
    (void)n_in; (void)out_size; (void)ws_size;
    const float* predo  = (const float*)d_in[0];
    const int*   target = (const int*)d_in[1];
    const int*   epoch  = (const int*)d_in[2];
    const int*   batch  = (const int*)d_in[3];
    const int    N      = in_sizes[1];          // rows; C is fixed at 256
    float*       out    = (float*)d_out;

    // workspace layout: meta(64B) | hist(4KB) | wbin(4KB) | partials(2KB) |
    //                   lp[N] f32 | bin[N] i32   (total ~16KB + 8N bytes)
    char*  ws       = (char*)d_ws;
    int*   meta     = (int*)ws;
    int*   hist     = (int*)(ws + 64);
    float* wbin     = (float*)(ws + 64 + 4096);
    float* partials = (float*)(ws + 64 + 8192);
    float* lp       = (float*)(ws + 16384);
    int*   bidx     = (int*)(ws + 16384 + (size_t)N * sizeof(float));

    ghmc_setup  <<<1,          256, 0, stream>>>(epoch, batch, meta, hist);
    ghmc_pass1  <<<P1_BLOCKS,  256, 0, stream>>>(predo, target, meta, hist, lp, bidx, N);
    ghmc_weights<<<1,          256, 0, stream>>>(hist, meta, wbin);
    ghmc_dot    <<<DOT_BLOCKS, 256, 0, stream>>>(lp, bidx, wbin, partials, N);
    ghmc_final  <<<1,           32, 0, stream>>>(partials, out);
}